// Top2ThresholdGating_3126736191786
// MI455X (gfx1250) — compile-verified
//
#include <hip/hip_runtime.h>
#include <hip/hip_bf16.h>
#include <math.h>

// ---------------------------------------------------------------------------
// Top-2 threshold gating (MoE router) for MI455X / gfx1250.
//   B=8, N=2048, D=1024, G=8 experts, CAP=320, out = [B,N,G,CAP] f32.
// Memory-bound: 64MB x-read + 168MB out-zero dominate (~10us roofline).
// Logits GEMM uses V_WMMA_F32_16X16X4_F32 (exact fp32, N padded 8->16).
// ---------------------------------------------------------------------------

typedef float v2f __attribute__((ext_vector_type(2)));
typedef float v8f __attribute__((ext_vector_type(8)));

#define BATCH   8
#define NTOK    2048
#define DMODEL  1024
#define NGATE   8
#define CAP     320            // max(min(2048, int(2048*1.25/8)), 4)
#define EPSG    1e-9f
#define THRESH  0.2f

// ---------------------------------------------------------------------------
// Kernel 1: zero the (mostly sparse) 168MB output with wide stores.
// ---------------------------------------------------------------------------
__global__ void zero_out_kernel(float4* __restrict__ p, int n4) {
    int i = blockIdx.x * blockDim.x + threadIdx.x;
    int stride = gridDim.x * blockDim.x;
    float4 z = make_float4(0.f, 0.f, 0.f, 0.f);
    for (; i < n4; i += stride) p[i] = z;
}

// ---------------------------------------------------------------------------
// Kernel 2: logits[row][g] = x[row][:] . W[:][g], row = b*NTOK + t.
// One wave (32 lanes) computes a 16x16 tile (16 tokens x 16 padded gates)
// via 256 chained v_wmma_f32_16x16x4_f32.
//
// A (16x4 f32) lane layout (ISA 7.12.2): lanes 0-15 hold M=lane, K={0,1};
//   lanes 16-31 hold M=lane-16, K={2,3}; 2 VGPRs (v2f).
// B (4x16 f32) mirrored: lanes 0-15 -> N=lane, K rows {0,1}; lanes 16-31 ->
//   N=lane-16, K rows {2,3}.  Gate columns 8..15 are zero-padded in LDS so
//   EXEC stays all-ones (WMMA requirement) with no branches in the K-loop.
// D (16x16 f32): VGPR v -> M = v + 8*(lane>=16), N = lane&15.
// ---------------------------------------------------------------------------
__global__ void __launch_bounds__(256)
logits_wmma_kernel(const float* __restrict__ x,
                   const float* __restrict__ w,
                   float* __restrict__ logits) {
    __shared__ float wlds[DMODEL * 16];   // 64KB: W padded to 16 columns

    int tid = threadIdx.x;
    for (int i = tid; i < DMODEL * 16; i += 256) {
        int r = i >> 4, c = i & 15;
        wlds[i] = (c < NGATE) ? w[r * NGATE + c] : 0.0f;
    }
    __syncthreads();

    int lane = tid & 31;
    int wv   = tid >> 5;
    int tile = blockIdx.x * 8 + wv;          // 1024 tiles total
    int r0   = tile << 4;                    // first of 16 token-rows
    int m    = lane & 15;
    int kh   = lane >> 4;                    // lane-half selects K pair
    int ncol = lane & 15;

    const float* xr = x + (size_t)(r0 + m) * DMODEL + kh * 2;

    v8f acc = {0.f, 0.f, 0.f, 0.f, 0.f, 0.f, 0.f, 0.f};

    for (int k0 = 0; k0 < DMODEL; k0 += 4) {
        v2f a = *(const v2f*)(xr + k0);               // global_load_b64
        int kb = (k0 + kh * 2) << 4;                  // row*16 in padded LDS
        v2f bfrag;
        bfrag.x = wlds[kb + ncol];
        bfrag.y = wlds[kb + 16 + ncol];
        acc = __builtin_amdgcn_wmma_f32_16x16x4_f32(
            false, a, false, bfrag, (short)0, acc, false, false);
    }

    if (ncol < NGATE) {
        int mb = r0 + kh * 8;
        #pragma unroll
        for (int v = 0; v < 8; ++v)
            logits[(size_t)(mb + v) * NGATE + ncol] = acc[v];
    }
}

// ---------------------------------------------------------------------------
// Packed per-expert 16-bit counters: 8 counters in unsigned c[4].
// Field e lives in c[e>>1], halfword (e&1). Max value 2048 < 65536, and
// packed 32-bit add/sub never carries across fields.
// ---------------------------------------------------------------------------
__device__ inline void fadd(unsigned c[4], int e, unsigned val) {
    unsigned inc = val << ((e & 1) * 16);
    #pragma unroll
    for (int q = 0; q < 4; ++q) c[q] += (q == (e >> 1)) ? inc : 0u;
}
__device__ inline unsigned fget(const unsigned c[4], int e) {
    unsigned r = 0;
    #pragma unroll
    for (int q = 0; q < 4; ++q) r |= (q == (e >> 1)) ? c[q] : 0u;
    return (r >> ((e & 1) * 16)) & 0xffffu;
}

// ---------------------------------------------------------------------------
// Kernel 3: per-batch gating + token-rank scan + sparse scatter.
// Grid = 8 (one block per batch), 256 threads, 8 tokens/thread.
// ---------------------------------------------------------------------------
__global__ void __launch_bounds__(256)
gating_scan_kernel(const float* __restrict__ logits, float* __restrict__ out) {
    __shared__ unsigned s1[256][4];
    __shared__ unsigned s2[256][4];

    const int bb  = blockIdx.x;        // batch
    const int tid = threadIdx.x;
    const int TPT = NTOK / 256;        // 8 tokens per thread

    float    g1s[TPT], g2s[TPT];
    unsigned meta[TPT];                // idx1 | idx2<<3 | keep<<6
    unsigned c1[4] = {0, 0, 0, 0};     // local top-1 counts
    unsigned c2[4] = {0, 0, 0, 0};     // local thresholded top-2 counts

    // ---- Phase A: per-token softmax / top1 / top2 / gate normalization ----
    #pragma unroll
    for (int i = 0; i < TPT; ++i) {
        int t = tid * TPT + i;
        const float* row = logits + ((size_t)bb * NTOK + t) * NGATE;
        float l[NGATE];
        float4 p0 = ((const float4*)row)[0];
        float4 p1 = ((const float4*)row)[1];
        l[0] = p0.x; l[1] = p0.y; l[2] = p0.z; l[3] = p0.w;
        l[4] = p1.x; l[5] = p1.y; l[6] = p1.z; l[7] = p1.w;

        float mx = l[0];
        #pragma unroll
        for (int e = 1; e < NGATE; ++e) mx = fmaxf(mx, l[e]);
        float ge[NGATE], s = 0.f;
        #pragma unroll
        for (int e = 0; e < NGATE; ++e) { ge[e] = expf(l[e] - mx); s += ge[e]; }
        float inv = 1.0f / s;

        int i1 = 0; float b1 = ge[0];                 // first-max (argmax)
        #pragma unroll
        for (int e = 1; e < NGATE; ++e) if (ge[e] > b1) { b1 = ge[e]; i1 = e; }
        int i2 = 0; float b2 = -1.0f;
        #pragma unroll
        for (int e = 0; e < NGATE; ++e)
            if (e != i1 && ge[e] > b2) { b2 = ge[e]; i2 = e; }

        float g1 = b1 * inv, g2 = b2 * inv;
        float g1n = g1 / (g1 + g2 + EPSG);
        float g2n = g2 / (g1n + g2 + EPSG);           // faithful: updated g1
        int keep = (g2n > THRESH) ? 1 : 0;

        g1s[i] = g1n; g2s[i] = g2n;
        meta[i] = (unsigned)i1 | ((unsigned)i2 << 3) | ((unsigned)keep << 6);
        fadd(c1, i1, 1u);
        if (keep) fadd(c2, i2, 1u);
    }

    // ---- Phase B: block-wide inclusive scan of packed count vectors ----
    unsigned v1[4], v2[4];
    #pragma unroll
    for (int q = 0; q < 4; ++q) {
        v1[q] = c1[q]; v2[q] = c2[q];
        s1[tid][q] = c1[q]; s2[tid][q] = c2[q];
    }
    for (int off = 1; off < 256; off <<= 1) {
        unsigned t1[4] = {0, 0, 0, 0}, t2[4] = {0, 0, 0, 0};
        __syncthreads();
        if (tid >= off) {
            #pragma unroll
            for (int q = 0; q < 4; ++q) { t1[q] = s1[tid - off][q]; t2[q] = s2[tid - off][q]; }
        }
        __syncthreads();
        if (tid >= off) {
            #pragma unroll
            for (int q = 0; q < 4; ++q) {
                v1[q] += t1[q]; v2[q] += t2[q];
                s1[tid][q] = v1[q]; s2[tid][q] = v2[q];
            }
        }
    }
    __syncthreads();

    unsigned m1cp[4];                                  // min(count1_total, CAP)
    unsigned r1[4], r2[4];
    #pragma unroll
    for (int q = 0; q < 4; ++q) {
        unsigned tot = s1[255][q];
        unsigned lo = tot & 0xffffu, hi = tot >> 16;
        lo = lo < CAP ? lo : CAP; hi = hi < CAP ? hi : CAP;
        m1cp[q] = lo | (hi << 16);
        r1[q] = v1[q] - c1[q];                         // exclusive prefix
        r2[q] = v2[q] - c2[q];
    }

    // ---- Phase C: replay tokens with running ranks; scatter nonzeros ----
    #pragma unroll
    for (int i = 0; i < TPT; ++i) {
        int t = tid * TPT + i;
        size_t rowbase = ((size_t)bb * NTOK + t) * NGATE;
        unsigned mt = meta[i];
        int e1 = mt & 7, e2 = (mt >> 3) & 7, kp = (mt >> 6) & 1;

        unsigned pos1 = fget(r1, e1); fadd(r1, e1, 1u);
        if (pos1 < CAP)
            out[(rowbase + e1) * CAP + pos1] = g1s[i];

        if (kp) {
            unsigned rk2 = fget(r2, e2); fadd(r2, e2, 1u);
            unsigned pos2 = rk2 + fget(m1cp, e2);
            if (pos2 < CAP)
                out[(rowbase + e2) * CAP + pos2] = g2s[i];
        }
    }
}

// ---------------------------------------------------------------------------
extern "C" void kernel_launch(void* const* d_in, const int* in_sizes, int n_in,
                              void* d_out, int out_size, void* d_ws, size_t ws_size,
                              hipStream_t stream) {
    const float* x = (const float*)d_in[0];   // [8,2048,1024] f32
    const float* w = (const float*)d_in[1];   // [1024,8] f32
    float* out    = (float*)d_out;            // [8,2048,8,320] f32
    float* logits = (float*)d_ws;             // 16384*8 f32 = 512 KB scratch

    // 1) zero the 168MB output (dominant bandwidth cost)
    int n4 = out_size / 4;
    zero_out_kernel<<<4096, 256, 0, stream>>>((float4*)out, n4);

    // 2) logits GEMM via fp32 WMMA: 1024 tiles, 8 waves/block
    logits_wmma_kernel<<<128, 256, 0, stream>>>(x, w, logits);

    // 3) per-batch gating + scan + scatter
    gating_scan_kernel<<<BATCH, 256, 0, stream>>>(logits, out);
}